// SymmetryAwareTaylorApproximatedAttention_72902774882841
// MI455X (gfx1250) — compile-verified
//
#include <hip/hip_runtime.h>

// MI455X (gfx1250) Taylor-attention kernel, round 2.
// - All matmuls on V_WMMA_F32_16X16X4_F32 (fp32 path: reference is fp32 and
//   cubes the scores, so low-precision inputs would wreck accuracy).
// - K/V tiles staged global->LDS by the Tensor Data Mover (tensor_load_to_lds)
//   with descriptor-side row padding (64 dwords data + 4 dwords pad = 68-float
//   stride) for bank-conflict-free WMMA fragment reads; double-buffered so the
//   next round's DMA overlaps this round's compute; synced with
//   s_wait_tensorcnt + workgroup barrier.
// - Causality restructured so the steady-state loop has NO masking and no
//   divergent branches: every round but the last is fully dense; the final
//   round does `wave` dense subtiles + one masked diagonal subtile, with
//   wave-uniform trip counts forced scalar via readfirstlane.

typedef __attribute__((ext_vector_type(2))) float v2f;
typedef __attribute__((ext_vector_type(8))) float v8f;
typedef __attribute__((ext_vector_type(4))) unsigned int u32x4;
typedef __attribute__((ext_vector_type(8))) int i32x8;
typedef __attribute__((ext_vector_type(4))) int i32x4;

#define SEQ   2048
#define DHEAD 64
#define KSTR  68   // LDS row stride (floats): 64 data + 4 TDM pad dwords
#define WSTR  18   // per-wave W scratch stride: 8B-aligned v2f reads, bank-spread

__device__ __forceinline__ v8f wmma_f32(v2f a, v2f b, v8f c) {
  // 8 args: (neg_a, A, neg_b, B, c_mod, C, reuse_a, reuse_b)
  return __builtin_amdgcn_wmma_f32_16x16x4_f32(false, a, false, b, (short)0, c,
                                               false, false);
}

// TDM: DMA a 64x64 fp32 tile (row stride 64 elems in memory) into LDS with
// 4-dword padding after every 64 dwords -> LDS row stride = KSTR floats.
__device__ __forceinline__ void tdm_load_tile_64x64(const float* gsrc,
                                                    unsigned lds_off) {
  unsigned long long ga = (unsigned long long)(uintptr_t)gsrc;
  u32x4 g0;
  g0[0] = 1u;                                    // count=1 (valid), user mode
  g0[1] = lds_off;                               // lds_addr [63:32]
  g0[2] = (unsigned)(ga & 0xffffffffu);          // global_addr [95:64]
  g0[3] = (unsigned)((ga >> 32) & 0x01ffffffu)   // global_addr [120:96]
          | 0x80000000u;                         // type=2 ("image") [127:126]
  i32x8 g1;
  g1[0] = (2 << 16)      // data_size = 4B
        | (1 << 20)      // pad_enable
        | (5 << 22)      // pad_interval: 64 dwords
        | (3 << 25);     // pad_amount : 4 dwords
  g1[1] = (int)(64u << 16);    // tensor_dim0 = 64     (bits 79:48)
  g1[2] = (int)(2048u << 16);  // tensor_dim1 = 2048   (bits 111:80)
  g1[3] = (int)(64u << 16);    // tile_dim0   = 64     (bits 127:112)
  g1[4] = 64;                  // tile_dim1   = 64     (bits 143:128)
  g1[5] = 64;                  // tensor_dim0_stride = 64 (bits 207:160)
  g1[6] = 0;
  g1[7] = 0;
  i32x4 gz = {0, 0, 0, 0};     // groups 2/3 unused (2D tile)
#if __clang_major__ >= 23
  i32x8 gz8 = {0, 0, 0, 0, 0, 0, 0, 0};
  __builtin_amdgcn_tensor_load_to_lds(g0, g1, gz, gz, gz8, 0);
#else
  __builtin_amdgcn_tensor_load_to_lds(g0, g1, gz, gz, 0);
#endif
}

__global__ __launch_bounds__(128) void taylor_attn_kernel(
    const float* __restrict__ Q, const float* __restrict__ K,
    const float* __restrict__ V, float* __restrict__ O) {
  __shared__ float Kl[2][64 * KSTR];
  __shared__ float Vl[2][64 * KSTR];
  __shared__ float Wl[4][16 * WSTR];

  const int tid  = threadIdx.x;
  const int lane = tid & 31;
  const int wave = tid >> 5;
  const int lo   = lane & 15;   // column / row index within half-wave
  const int hi   = lane >> 4;   // half-wave select
  // Wave-uniform copy the compiler can prove uniform -> scalar branches.
  const int wv = __builtin_amdgcn_readfirstlane(wave);

  const int bh = blockIdx.y;
  const int qb = blockIdx.x * 64;
  const int q0 = qb + wave * 16;      // this wave's query-tile start row

  const float* Qp = Q + (size_t)bh * SEQ * DHEAD;
  const float* Kp = K + (size_t)bh * SEQ * DHEAD;
  const float* Vp = V + (size_t)bh * SEQ * DHEAD;
  float*       Op = O + (size_t)bh * SEQ * DHEAD;

  // Q tile (16x64) in WMMA A-layout fragments: frag f covers d = 4f..4f+3;
  // lane holds (row = lo, d = 4f + 2*hi + {0,1}).
  v2f qa[16];
  {
    const float* qrow = Qp + (size_t)(q0 + lo) * DHEAD + 2 * hi;
#pragma unroll
    for (int f = 0; f < 16; ++f) qa[f] = *(const v2f*)(qrow + 4 * f);
  }

  v8f o0 = {}, o1 = {}, o2 = {}, o3 = {};   // O tile 16x64 (C-layout)
  float z[8];
#pragma unroll
  for (int r = 0; r < 8; ++r) z[r] = 0.f;

  // One 16x16 key subtile: S = Q*K^T (16 WMMAs), Taylor weights (+optional
  // diagonal mask), Z partials, C->A relayout via per-wave LDS scratch
  // (same-wave DS ops retire in order), then O += W*V (16 WMMAs).
  auto subtile = [&](const float* Kb, const float* Vb, int u, bool diag) {
    v8f s = {};
    const float* kl = Kb + (u * 16 + lo) * KSTR + 2 * hi;
#pragma unroll
    for (int f = 0; f < 16; ++f) {
      v2f bf = *(const v2f*)(kl + 4 * f);
      s = wmma_f32(qa[f], bf, s);
    }
    float* wscr = &Wl[wave][0];
#pragma unroll
    for (int r = 0; r < 8; ++r) {
      int   row = r + 8 * hi;             // C-layout: row = r + 8*hi, col = lo
      float sv  = s[r];
      float wt  = 1.f + sv * (1.f + sv * (0.5f + sv * 0.16666666666666666f));
      if (diag) wt = (row < lo) ? 0.f : wt;
      z[r] += wt;
      wscr[row * WSTR + lo] = wt;
    }
#pragma unroll
    for (int kk = 0; kk < 4; ++kk) {
      v2f af = *(const v2f*)&Wl[wave][lo * WSTR + 4 * kk + 2 * hi];
      const float* vb = Vb + (u * 16 + 4 * kk + 2 * hi) * KSTR + lo;
      v2f b;
      b[0] = vb[0];   b[1] = vb[KSTR];       o0 = wmma_f32(af, b, o0);
      b[0] = vb[16];  b[1] = vb[KSTR + 16];  o1 = wmma_f32(af, b, o1);
      b[0] = vb[32];  b[1] = vb[KSTR + 32];  o2 = wmma_f32(af, b, o2);
      b[0] = vb[48];  b[1] = vb[KSTR + 48];  o3 = wmma_f32(af, b, o3);
    }
  };

  const int nrounds = blockIdx.x + 1;       // causal: keys 0 .. qb+63

  // Lead wave primes round 0 via TDM. Scalar (wave-uniform) branch: TDM
  // ignores EXEC, so it must be branched over, not exec-masked.
  if (wv == 0) {
    tdm_load_tile_64x64(Kp, (unsigned)(uintptr_t)&Kl[0][0]);
    tdm_load_tile_64x64(Vp, (unsigned)(uintptr_t)&Vl[0][0]);
  }

  for (int t = 0; t < nrounds; ++t) {
    const int cur = t & 1;
    if (wv == 0) {
      if (t + 1 < nrounds) {
        // Issue next round into the other buffer (free since the end-of-round
        // barrier of t-1), then wait for the current round's two loads: with
        // in-order TENSORcnt retire, <=2 outstanding means round t is done.
        tdm_load_tile_64x64(Kp + (size_t)(t + 1) * 64 * DHEAD,
                            (unsigned)(uintptr_t)&Kl[cur ^ 1][0]);
        tdm_load_tile_64x64(Vp + (size_t)(t + 1) * 64 * DHEAD,
                            (unsigned)(uintptr_t)&Vl[cur ^ 1][0]);
        __builtin_amdgcn_s_wait_tensorcnt((short)2);
      } else {
        __builtin_amdgcn_s_wait_tensorcnt((short)0);
      }
    }
    __syncthreads();                       // publish LDS tile to all waves

    const float* Kb = &Kl[cur][0];
    const float* Vb = &Vl[cur][0];
    if (t + 1 < nrounds) {
      // Steady state: all 4 subtiles dense, zero masking, zero branching.
#pragma unroll
      for (int u = 0; u < 4; ++u) subtile(Kb, Vb, u, false);
    } else {
      // Final round: wave w has `w` dense subtiles + the diagonal subtile.
      for (int u = 0; u < wv; ++u) subtile(Kb, Vb, u, false);
      subtile(Kb, Vb, wv, true);
    }
    __syncthreads();                       // buffer reuse fence
  }

  // Z: reduce each row's partials across the 16 lanes of each half-wave.
#pragma unroll
  for (int r = 0; r < 8; ++r) {
    float zv = z[r];
    zv += __shfl_xor(zv, 1, 32);
    zv += __shfl_xor(zv, 2, 32);
    zv += __shfl_xor(zv, 4, 32);
    zv += __shfl_xor(zv, 8, 32);
    float* orow = Op + (size_t)(q0 + r + 8 * hi) * DHEAD + lo;
    float vals[4] = {o0[r], o1[r], o2[r], o3[r]};
#pragma unroll
    for (int nt = 0; nt < 4; ++nt) {
      float rv = vals[nt] / zv;             // nan_to_num semantics
      if (__builtin_isnan(rv))       rv = 0.f;
      else if (__builtin_isinf(rv))  rv = rv > 0.f ? 3.402823466e+38f
                                                   : -3.402823466e+38f;
      orow[nt * 16] = rv;
    }
  }
}

extern "C" void kernel_launch(void* const* d_in, const int* in_sizes, int n_in,
                              void* d_out, int out_size, void* d_ws, size_t ws_size,
                              hipStream_t stream) {
  const float* Q = (const float*)d_in[0];
  const float* K = (const float*)d_in[1];
  const float* V = (const float*)d_in[2];
  float*       O = (float*)d_out;
  (void)n_in; (void)out_size; (void)d_ws; (void)ws_size;

  const int BH = in_sizes[0] / (SEQ * DHEAD);   // B*H = 16
  dim3 grid(SEQ / 64, BH);
  taylor_attn_kernel<<<grid, dim3(128), 0, stream>>>(Q, K, V, O);
}